// ImageOnlyDecomposer_3856880631987
// MI455X (gfx1250) — compile-verified
//
#include <hip/hip_runtime.h>
#include <math.h>

// ---------------------------------------------------------------------------
// MI455X (gfx1250, wave32) implementation.
// Pipeline: qkv GEMM -> attention -> outproj(+resid) GEMM -> w1 GEMM ->
//           LN+GELU -> w2 GEMM -> expand/topk/normalize.
// All GEMMs use V_WMMA_F32_16X16X4_F32 (exact fp32 math).
// All out-of-range handling is done with ADDRESS CLAMPS (no data selects),
// so every inner loop is branch-free and loads can be pipelined vs WMMA.
// ---------------------------------------------------------------------------

typedef float v2f __attribute__((ext_vector_type(2)));
typedef float v8f __attribute__((ext_vector_type(8)));

constexpr int Bb   = 16;
constexpr int Nn   = 196;
constexpr int Dd   = 512;
constexpr int Mm   = 20;
constexpr int Hh   = 8;
constexpr int HDim = 64;
constexpr int ROWS = Bb * Nn;   // 3136 (multiple of 16)
constexpr int NP   = 208;       // 13 * 16, padded sequence length

#define DEV __device__ __forceinline__

DEV v8f wmma_f32(v2f a, v2f b, v8f c) {
  // D = A(16x4) * B(4x16) + C(16x16), fp32.
  return __builtin_amdgcn_wmma_f32_16x16x4_f32(false, a, false, b, (short)0, c,
                                               false, false);
}

DEV float wredf(float x) {
  for (int o = 16; o; o >>= 1) x += __shfl_xor(x, o, 32);
  return x;
}
DEV int wredi(int x) {
  for (int o = 16; o; o >>= 1) x += __shfl_xor(x, o, 32);
  return x;
}

// ---------------------------------------------------------------------------
// Generic row-tile GEMM: out[row,col] = sum_k A[row,k] * W[col,k] + bias[col]
//                         (+ resid[row,col] when MODE==1)
// A: [ROWS x 512] row-major, W: [ncols x 512] row-major.
// Block: 128 threads (4 waves). Each block: 16 rows x 64 cols.
// ---------------------------------------------------------------------------
template <int MODE>
__global__ __launch_bounds__(128) void gemm_rt(
    const float* __restrict__ A, const float* __restrict__ W,
    const float* __restrict__ bias, const float* __restrict__ resid,
    float* __restrict__ out, int ncols, int ldo) {
  __shared__ float Alds[16 * Dd];  // 32 KB contiguous A stripe

  const float* Ablk = A + (size_t)blockIdx.x * (16 * Dd);
  for (int i = threadIdx.x; i < (16 * Dd) / 4; i += 128)
    ((float4*)Alds)[i] = ((const float4*)Ablk)[i];

  const int wave = threadIdx.x >> 5;
  const int lane = threadIdx.x & 31;
  const int lo = lane & 15;   // M (from LDS) / N (from W) index within tile
  const int hi = lane >> 4;   // K half-select

  const int col = blockIdx.y * 64 + wave * 16 + lo;
  const int colLoad = (col < ncols) ? col : 0;  // address clamp only
  const float* Wc = W + (size_t)colLoad * Dd;

  // Warm the cache path for this lane's weight row (global_prefetch_b8).
  __builtin_prefetch(Wc, 0, 1);
  __builtin_prefetch(Wc + 256, 0, 1);

  __syncthreads();
  const float* Ar = &Alds[lo * Dd];

  v8f acc = {};
#pragma unroll 8
  for (int kk = 0; kk < Dd; kk += 4) {
    v2f a = *(const v2f*)&Ar[kk + 2 * hi];  // A[m][kk+2hi .. +1]
    v2f b = *(const v2f*)&Wc[kk + 2 * hi];  // B[kk+2hi..][n] = W[n][k]
    acc = wmma_f32(a, b, acc);
  }

  if (col < ncols) {
    const float bv = bias[col];
#pragma unroll
    for (int r = 0; r < 8; ++r) {
      const int row = blockIdx.x * 16 + r + 8 * hi;  // D-matrix row layout
      float v = acc[r] + bv;
      if (MODE == 1) v += resid[(size_t)row * ldo + col];
      out[(size_t)row * ldo + col] = v;
    }
  }
}

// ---------------------------------------------------------------------------
// Attention: one block per (row-stripe s, head h, batch b); 128 threads.
// qkv layout: [ROWS x 1536]; q at +h*64, k at +512+h*64, v at +1024+h*64.
// Out-of-range rows/cols use clamped addresses: garbage S columns >=196 are
// zeroed in softmax; garbage S rows >=196 are never stored; clamped v rows
// are multiplied by zeroed P columns. No data masking needed anywhere.
// ---------------------------------------------------------------------------
__global__ __launch_bounds__(128) void attn_kernel(
    const float* __restrict__ qkv, float* __restrict__ o) {
  __shared__ float S[16 * NP];  // 13 KB

  const int s = blockIdx.x, h = blockIdx.y, b = blockIdx.z;
  const int wave = threadIdx.x >> 5;
  const int lane = threadIdx.x & 31;
  const int lo = lane & 15, hi = lane >> 4;

  const size_t base = (size_t)b * Nn * 1536 + h * 64;

  // ---- Phase 1: S = (q k^T)/8 ----
  const int nq = min(s * 16 + lo, Nn - 1);             // address clamp
  const float* qrow = qkv + base + (size_t)nq * 1536;
  for (int t = wave; t < 13; t += 4) {
    const int nk = min(t * 16 + lo, Nn - 1);           // address clamp
    const float* krow = qkv + base + 512 + (size_t)nk * 1536;
    v8f acc = {};
#pragma unroll
    for (int kk = 0; kk < HDim; kk += 4) {
      v2f a = *(const v2f*)&qrow[kk + 2 * hi];
      v2f bb = *(const v2f*)&krow[kk + 2 * hi];
      acc = wmma_f32(a, bb, acc);
    }
#pragma unroll
    for (int r = 0; r < 8; ++r)
      S[(r + 8 * hi) * NP + t * 16 + lo] = acc[r] * 0.125f;  // 1/sqrt(64)
  }
  __syncthreads();

  // ---- Phase 2: softmax (8 lanes per row, 16 rows, 128 threads) ----
  {
    const int row = threadIdx.x >> 3, sub = threadIdx.x & 7;
    float* Sr = &S[row * NP];
    float mx = -3.0e38f;
    for (int c = sub; c < Nn; c += 8) mx = fmaxf(mx, Sr[c]);
    for (int off = 4; off; off >>= 1) mx = fmaxf(mx, __shfl_xor(mx, off, 8));
    float sum = 0.0f;
    for (int c = sub; c < Nn; c += 8) {
      float e = expf(Sr[c] - mx);
      Sr[c] = e;
      sum += e;
    }
    for (int off = 4; off; off >>= 1) sum += __shfl_xor(sum, off, 8);
    const float inv = 1.0f / sum;
    for (int c = sub; c < Nn; c += 8) Sr[c] *= inv;
    for (int c = Nn + sub; c < NP; c += 8) Sr[c] = 0.0f;  // mask padding cols
  }
  __syncthreads();

  // ---- Phase 3: O = P @ v (wave -> one 16-col tile of hd=64) ----
  {
    v8f acc = {};
    const float* vbase = qkv + base + 1024 + wave * 16 + lo;
    // Main body: all v rows valid (kk+2hi+1 <= 191+2+1 < 196). Branch-free.
#pragma unroll 4
    for (int kk = 0; kk < 192; kk += 4) {
      v2f a = *(const v2f*)&S[lo * NP + kk + 2 * hi];
      const int r0 = kk + 2 * hi;
      v2f bb;
      bb.x = vbase[(size_t)r0 * 1536];
      bb.y = vbase[(size_t)(r0 + 1) * 1536];
      acc = wmma_f32(a, bb, acc);
    }
    // Tail: clamp v row addresses; P columns >=196 are zero so clamped
    // values contribute nothing.
#pragma unroll
    for (int kk = 192; kk < NP; kk += 4) {
      v2f a = *(const v2f*)&S[lo * NP + kk + 2 * hi];
      const int r0 = min(kk + 2 * hi, Nn - 1);
      const int r1 = min(kk + 2 * hi + 1, Nn - 1);
      v2f bb;
      bb.x = vbase[(size_t)r0 * 1536];
      bb.y = vbase[(size_t)r1 * 1536];
      acc = wmma_f32(a, bb, acc);
    }
#pragma unroll
    for (int r = 0; r < 8; ++r) {
      const int row = s * 16 + r + 8 * hi;
      if (row < Nn)
        o[((size_t)b * Nn + row) * Dd + h * 64 + wave * 16 + lo] = acc[r];
    }
  }
}

// ---------------------------------------------------------------------------
// LayerNorm (biased var) + exact GELU. One wave per row of 512.
// ---------------------------------------------------------------------------
__global__ __launch_bounds__(256) void ln_gelu_kernel(
    const float* __restrict__ h, const float* __restrict__ g,
    const float* __restrict__ beta, float* __restrict__ out) {
  const int wave = threadIdx.x >> 5, lane = threadIdx.x & 31;
  const int row = blockIdx.x * 8 + wave;
  const float* hr = h + (size_t)row * Dd + lane * 16;

  float x[16];
#pragma unroll
  for (int jv = 0; jv < 4; ++jv) {
    float4 f = ((const float4*)hr)[jv];
    x[4 * jv + 0] = f.x;
    x[4 * jv + 1] = f.y;
    x[4 * jv + 2] = f.z;
    x[4 * jv + 3] = f.w;
  }
  float sum = 0.0f, sq = 0.0f;
#pragma unroll
  for (int j = 0; j < 16; ++j) {
    sum += x[j];
    sq += x[j] * x[j];
  }
  sum = wredf(sum);
  sq = wredf(sq);
  const float mu = sum * (1.0f / Dd);
  const float var = sq * (1.0f / Dd) - mu * mu;
  const float rstd = rsqrtf(var + 1e-5f);

  const float* gr = g + lane * 16;
  const float* br = beta + lane * 16;
  float* orow = out + (size_t)row * Dd + lane * 16;
#pragma unroll
  for (int jv = 0; jv < 4; ++jv) {
    float4 gv = ((const float4*)gr)[jv];
    float4 bv = ((const float4*)br)[jv];
    float4 y;
    float* yp = &y.x;
    const float* gp = &gv.x;
    const float* bp = &bv.x;
#pragma unroll
    for (int q = 0; q < 4; ++q) {
      float t = (x[4 * jv + q] - mu) * rstd * gp[q] + bp[q];
      yp[q] = 0.5f * t * (1.0f + erff(t * 0.70710678118654752f));
    }
    ((float4*)orow)[jv] = y;
  }
}

// ---------------------------------------------------------------------------
// Expansion + abs-top-k(k) over D + L2 normalize. One block per (b,n).
// a[m][d] = F[b,n,d] * templates[m,d] * Qp[b,n,m]
// Top-k threshold: radix/binary search on IEEE bits of |a| (wave-level),
// ties broken by ascending d (matches jax.lax.top_k). LDS-staged output,
// flushed coalesced into [B,N,D,M].
// ---------------------------------------------------------------------------
__global__ __launch_bounds__(256) void expand_topk_kernel(
    const float* __restrict__ F, const float* __restrict__ templ,
    const float* __restrict__ Qp, const int* __restrict__ kptr,
    float* __restrict__ out) {
  __shared__ float Fsh[Dd];
  __shared__ float Qsh[Mm];
  __shared__ float Osh[Mm * Dd];  // [m][d], 40 KB

  const int bn = blockIdx.x;
  const int tid = threadIdx.x, wave = tid >> 5, lane = tid & 31;
  const int kval = *kptr;

  if (tid < Dd / 4) ((float4*)Fsh)[tid] = ((const float4*)(F + (size_t)bn * Dd))[tid];
  if (tid < Mm) Qsh[tid] = Qp[(size_t)bn * Mm + tid];
  __syncthreads();

  for (int m = wave; m < Mm; m += 8) {
    const float s = Qsh[m];
    const float* tr = templ + (size_t)m * Dd + lane * 16;

    float a[16];
    unsigned u[16];
#pragma unroll
    for (int jv = 0; jv < 4; ++jv) {
      float4 t4 = ((const float4*)tr)[jv];
      const float* fp = &Fsh[lane * 16 + 4 * jv];
      a[4 * jv + 0] = fp[0] * t4.x * s;
      a[4 * jv + 1] = fp[1] * t4.y * s;
      a[4 * jv + 2] = fp[2] * t4.z * s;
      a[4 * jv + 3] = fp[3] * t4.w * s;
    }
#pragma unroll
    for (int j = 0; j < 16; ++j) u[j] = __float_as_uint(fabsf(a[j]));

    // kth-largest bit pattern: largest thr with count(u >= thr) >= k
    unsigned thr = 0;
    for (int bit = 30; bit >= 0; --bit) {
      const unsigned cand = thr | (1u << bit);
      int c = 0;
#pragma unroll
      for (int j = 0; j < 16; ++j) c += (u[j] >= cand) ? 1 : 0;
      if (wredi(c) >= kval) thr = cand;
    }

    int cgt = 0, ceq = 0;
#pragma unroll
    for (int j = 0; j < 16; ++j) {
      cgt += (u[j] > thr) ? 1 : 0;
      ceq += (u[j] == thr) ? 1 : 0;
    }
    cgt = wredi(cgt);

    // exclusive scan of per-lane equal-counts (index order d = lane*16 + j)
    int scan = ceq;
    for (int off = 1; off < 32; off <<= 1) {
      int y = __shfl_up(scan, off, 32);
      if (lane >= off) scan += y;
    }
    const int eqBase = scan - ceq;
    const int quota = kval - cgt;

    float ssq = 0.0f;
    int locEq = 0;
    unsigned selMask = 0;
#pragma unroll
    for (int j = 0; j < 16; ++j) {
      const bool eq = (u[j] == thr);
      const bool take = (u[j] > thr) || (eq && (eqBase + locEq) < quota);
      locEq += eq ? 1 : 0;
      if (take) {
        ssq += a[j] * a[j];
        selMask |= (1u << j);
      }
    }
    ssq = wredf(ssq);
    const float inv = 1.0f / fmaxf(sqrtf(ssq), 1e-6f);

    float* orow = &Osh[m * Dd + lane * 16];
#pragma unroll
    for (int j = 0; j < 16; ++j)
      orow[j] = ((selMask >> j) & 1u) ? a[j] * inv : 0.0f;
  }
  __syncthreads();

  // Coalesced flush: out[b,n,d,m] contiguous block of D*M per (b,n).
  float* dst = out + (size_t)bn * (Dd * Mm);
  for (int g2 = tid; g2 < Dd * Mm; g2 += 256) {
    const int d = g2 / Mm;
    const int m = g2 - d * Mm;
    dst[g2] = Osh[m * Dd + d];
  }
}

// ---------------------------------------------------------------------------
extern "C" void kernel_launch(void* const* d_in, const int* in_sizes, int n_in,
                              void* d_out, int out_size, void* d_ws,
                              size_t ws_size, hipStream_t stream) {
  (void)in_sizes; (void)n_in; (void)out_size; (void)ws_size;

  const float* F    = (const float*)d_in[0];   // [B,N,D]
  const float* Wqkv = (const float*)d_in[1];   // [3D,D]
  const float* bqkv = (const float*)d_in[2];   // [3D]
  const float* Wout = (const float*)d_in[3];   // [D,D]
  const float* bout = (const float*)d_in[4];   // [D]
  const float* W1   = (const float*)d_in[5];   // [D,D]
  const float* b1   = (const float*)d_in[6];   // [D]
  const float* lng  = (const float*)d_in[7];   // [D]
  const float* lnb  = (const float*)d_in[8];   // [D]
  const float* W2   = (const float*)d_in[9];   // [M,D]
  const float* b2   = (const float*)d_in[10];  // [M]
  const float* tpl  = (const float*)d_in[11];  // [M,D]
  const int*   kptr = (const int*)d_in[12];    // scalar k (=51)
  float* out = (float*)d_out;

  // Workspace layout (floats), with sequential buffer reuse (~32.1 MB total):
  float* ws  = (float*)d_ws;
  float* qkv = ws;                              // ROWS*1536
  float* o   = qkv + (size_t)ROWS * 1536;       // ROWS*512
  float* Fe  = o + (size_t)ROWS * Dd;           // ROWS*512
  float* h   = qkv;  // reuse (qkv dead after attention)
  float* hg  = o;    // reuse (o dead after outproj)
  float* Qpb = Fe;   // reuse (Fe dead after w1 gemm)

  const dim3 blk(128);
  // 1) qkv = F @ Wqkv^T + b
  gemm_rt<0><<<dim3(ROWS / 16, 1536 / 64), blk, 0, stream>>>(
      F, Wqkv, bqkv, nullptr, qkv, 1536, 1536);
  // 2) attention -> o [ROWS, D]
  attn_kernel<<<dim3(13, Hh, Bb), blk, 0, stream>>>(qkv, o);
  // 3) Fe = o @ Wout^T + bout + F
  gemm_rt<1><<<dim3(ROWS / 16, Dd / 64), blk, 0, stream>>>(
      o, Wout, bout, F, Fe, Dd, Dd);
  // 4) h = Fe @ W1^T + b1
  gemm_rt<0><<<dim3(ROWS / 16, Dd / 64), blk, 0, stream>>>(
      Fe, W1, b1, nullptr, h, Dd, Dd);
  // 5) hg = gelu(layernorm(h))
  ln_gelu_kernel<<<dim3(ROWS / 8), dim3(256), 0, stream>>>(h, lng, lnb, hg);
  // 6) Qp = hg @ W2^T + b2   [ROWS, 20]
  gemm_rt<0><<<dim3(ROWS / 16, 1), blk, 0, stream>>>(
      hg, W2, b2, nullptr, Qpb, Mm, Mm);
  // 7) expansion + top-k + normalize -> out [B,N,D,M]
  expand_topk_kernel<<<dim3(ROWS), dim3(256), 0, stream>>>(
      F, tpl, Qpb, kptr, out);
}